// EnhancedLesionPenaltyLoss_22050362098212
// MI455X (gfx1250) — compile-verified
//
#include <hip/hip_runtime.h>
#include <hip/hip_bf16.h>

// ---------------------------------------------------------------------------
// EnhancedLesionPenaltyLoss for MI455X (gfx1250, wave32)
// Pass 1: branchless streaming pass computing per-block sufficient statistics
//         (HBM-bound: 134 MB unique / 23.3 TB/s ~ 5.8 us; neighbor reads hit
//         L2 since the whole 134 MB tensor fits in the 192 MB L2).
// Pass 2: tiny finalize kernel; the 16-way batch reduction runs on the WMMA
//         pipe (v_wmma_f32_16x16x4_f32).
// ---------------------------------------------------------------------------

typedef __attribute__((ext_vector_type(2))) float v2f;
typedef __attribute__((ext_vector_type(8))) float v8f;

#define LP_MIN_T     0.01f
#define LP_MAX_T     0.5f
#define LP_LESION_T  0.3f
#define LP_NVOX      2097152.0f      // 128^3
#define LP_DIFF_DEN  6242304.0f     // 3 * 127*128*128

#define BLOCKS_PER_BATCH 128
#define THREADS          256
#define ITERS            16          // (128^3/4 chunks) / (128*256 threads)

__global__ __launch_bounds__(THREADS)
void lesion_partial_kernel(const float* __restrict__ pred,
                           float* __restrict__ partials)
{
    const int b = blockIdx.y;
    const float*  base = pred + (size_t)b * (1u << 21);
    const float4* p4   = reinterpret_cast<const float4*>(base);

    float cnt = 0.f, nh = 0.f, s1 = 0.f, s2 = 0.f, gs = 0.f, vmax = -1.f;

#pragma unroll 4
    for (int it = 0; it < ITERS; ++it) {
        const int chunk = it * (BLOCKS_PER_BATCH * THREADS)
                        + blockIdx.x * THREADS + threadIdx.x;
        const int e  = chunk << 2;          // element index within batch
        const int d  = e >> 14;             // depth  (0..127)
        const int h  = (e >> 7) & 127;      // height (0..127)
        const int w0 = e & 127;             // width of first elem (mult of 4)

        // Branchless neighbor access: clamp offset to 0 when out of range
        // (re-reads own chunk from L0), mask the contribution with 0/1.
        const float wm = (w0 < 124) ? 1.f : 0.f;
        const float hm = (h  < 127) ? 1.f : 0.f;
        const float dm = (d  < 127) ? 1.f : 0.f;
        const int   wOff = (w0 < 124) ? 4    : 0;   // +1 in w (scalar)
        const int   hOff = (h  < 127) ? 32   : 0;   // +1 in h (float4 chunks)
        const int   dOff = (d  < 127) ? 4096 : 0;   // +1 in d (float4 chunks)

        // All four loads issue unconditionally -> one contiguous VMEM clause.
        const float4 ov = p4[chunk];
        const float4 hv = p4[chunk + hOff];
        const float4 dv = p4[chunk + dOff];
        const float  wv = base[e + wOff];

        if (it + 1 < ITERS)
            __builtin_prefetch(&p4[chunk + BLOCKS_PER_BATCH * THREADS], 0, 0);

        const float a0 = ov.x, a1 = ov.y, a2 = ov.z, a3 = ov.w;

#define LP_ACC(v) do {                                  \
            const float _v = (v);                        \
            const bool  _m = _v > LP_MIN_T;              \
            cnt += _m ? 1.f : 0.f;                       \
            s1  += _m ? _v  : 0.f;                       \
            s2  += _m ? _v * _v : 0.f;                   \
            nh  += (_v > LP_MAX_T) ? 1.f : 0.f;          \
            vmax = fmaxf(vmax, _v);                      \
        } while (0)

        LP_ACC(a0); LP_ACC(a1); LP_ACC(a2); LP_ACC(a3);
#undef LP_ACC

        // w-direction: 3 intra-chunk pairs always valid (w0 <= 124)
        gs += fabsf(a1 - a0) + fabsf(a2 - a1) + fabsf(a3 - a2);
        gs += wm * fabsf(wv - a3);
        // h-direction diffs (stride 128 floats = 32 chunks)
        gs += hm * (fabsf(hv.x - a0) + fabsf(hv.y - a1)
                  + fabsf(hv.z - a2) + fabsf(hv.w - a3));
        // d-direction diffs (stride 16384 floats = 4096 chunks)
        gs += dm * (fabsf(dv.x - a0) + fabsf(dv.y - a1)
                  + fabsf(dv.z - a2) + fabsf(dv.w - a3));
    }

    float lc = (vmax > LP_LESION_T) ? 1.f : 0.f;

    // fixed-order wave32 reduction (deterministic)
    for (int off = 16; off > 0; off >>= 1) {
        cnt += __shfl_down(cnt, off, 32);
        nh  += __shfl_down(nh,  off, 32);
        s1  += __shfl_down(s1,  off, 32);
        s2  += __shfl_down(s2,  off, 32);
        gs  += __shfl_down(gs,  off, 32);
        lc  += __shfl_down(lc,  off, 32);
    }

    __shared__ float red[THREADS / 32][6];
    const int wave = threadIdx.x >> 5;
    const int lane = threadIdx.x & 31;
    if (lane == 0) {
        red[wave][0] = cnt; red[wave][1] = nh; red[wave][2] = s1;
        red[wave][3] = s2;  red[wave][4] = gs; red[wave][5] = lc;
    }
    __syncthreads();
    if (threadIdx.x < 6) {
        float t = 0.f;
        for (int w = 0; w < THREADS / 32; ++w)   // fixed order
            t += red[w][threadIdx.x];
        partials[((size_t)b * BLOCKS_PER_BATCH + blockIdx.x) * 6 + threadIdx.x] = t;
    }
}

__global__ __launch_bounds__(256)
void lesion_final_kernel(const float* __restrict__ partials,
                         float* __restrict__ out)
{
    __shared__ float part[16][8][6];
    __shared__ float lossb[16];
    const int t = threadIdx.x;

    // Stage A: 128 threads pre-reduce 16 partials each (fixed order)
    if (t < 128) {
        const int b = t >> 3, g = t & 7;
        float acc[6] = {0.f, 0.f, 0.f, 0.f, 0.f, 0.f};
        for (int j = 0; j < 16; ++j) {
            const float* p = partials + ((size_t)b * BLOCKS_PER_BATCH + g * 16 + j) * 6;
            for (int c = 0; c < 6; ++c) acc[c] += p[c];
        }
        for (int c = 0; c < 6; ++c) part[b][g][c] = acc[c];
    }
    __syncthreads();

    // Stage B: one thread per batch computes its loss
    if (t < 16) {
        float cnt = 0.f, nh = 0.f, s1 = 0.f, s2 = 0.f, gs = 0.f, lc = 0.f;
        for (int g = 0; g < 8; ++g) {           // fixed order
            cnt += part[t][g][0]; nh += part[t][g][1]; s1 += part[t][g][2];
            s2  += part[t][g][3]; gs += part[t][g][4]; lc += part[t][g][5];
        }
        const float act  = cnt / LP_NVOX;
        const float high = nh  / LP_NVOX;
        float loss = fmaxf(0.005f - act, 0.f) * 15.f
                   + fmaxf(high - 0.03f, 0.f) * 5.f;
        const float avg_grad = gs * (1.f / LP_DIFF_DEN);
        if (lc > 0.f)                              // has_lesion
            loss += fminf(avg_grad, 1.f) * 5.f;
        const bool gate = (act > 0.001f) && (cnt > 1.f);
        if (gate) {
            const float m   = s1 / cnt;
            const float sq  = s2 - s1 * s1 / cnt;
            const float var = sq / fmaxf(cnt - 1.f, 1.f);
            const float sd  = sqrtf(fmaxf(var, 0.f));
            const float rel = sd / (m + 1e-6f);
            loss += expf(-5.f * rel) * 7.f;
        }
        lossb[t] = loss;
    }
    __syncthreads();

    // Stage C: 16-way batch reduction on the WMMA pipe.
    // A is 16x4 f32: lanes 0-15 VGPR0 hold A[m][0] = loss[m]; everything else 0.
    // B = ones(4x16)  =>  D[m][n] = loss[m] replicated across n.
    // Executed unconditionally by the whole block so EXEC is all-ones.
    const int lane = t & 31;
    v2f a;  a.x = (lane < 16) ? lossb[lane] : 0.f;  a.y = 0.f;
    v2f bm; bm.x = 1.f; bm.y = 1.f;
    v8f c = {};
    v8f dres = __builtin_amdgcn_wmma_f32_16x16x4_f32(
        /*neg_a=*/false, a, /*neg_b=*/false, bm,
        /*c_mod=*/(short)0, c, /*reuse_a=*/false, /*reuse_b=*/false);

    // lane n<16 holds D[0..7][n] -> sum_{m=0..7}; lane n>=16 -> sum_{m=8..15}
    float s = dres[0] + dres[1] + dres[2] + dres[3]
            + dres[4] + dres[5] + dres[6] + dres[7];
    const float total = s + __shfl(s, lane ^ 16, 32);

    if (t == 0)
        out[0] = total * (1.f / 16.f);
}

extern "C" void kernel_launch(void* const* d_in, const int* in_sizes, int n_in,
                              void* d_out, int out_size, void* d_ws, size_t ws_size,
                              hipStream_t stream)
{
    const float* pred     = (const float*)d_in[0];
    float*       out      = (float*)d_out;
    float*       partials = (float*)d_ws;   // 16 * 128 * 6 floats = 48 KB

    dim3 grid(BLOCKS_PER_BATCH, 16);
    lesion_partial_kernel<<<grid, THREADS, 0, stream>>>(pred, partials);
    lesion_final_kernel<<<1, 256, 0, stream>>>(partials, out);
}